// BatchGlobalPooling_8280696947332
// MI455X (gfx1250) — compile-verified
//
#include <hip/hip_runtime.h>
#include <stdint.h>

// ---------------------------------------------------------------------------
// Segment-mean pooling (sorted segments), MI455X / gfx1250.
//   out[g, f] = mean over nodes n with batch[n]==g of x[n, f]   (F = 64)
// Memory-bound: 512MB stream -> ~22us floor at 23.3 TB/s.
// Data path: global_load_async_to_lds_b128/b32 (ASYNCcnt pipeline, depth 6,
//            features AND batch ids staged ahead; no exposed load latency).
// Compute:   V_WMMA_F32_16X16X4_F32 with one-hot selection A-matrix over a
//            16-graph window; atomic flush on window change (zeros skipped).
// ---------------------------------------------------------------------------

typedef float v2f __attribute__((ext_vector_type(2)));
typedef float v8f __attribute__((ext_vector_type(8)));
typedef __attribute__((address_space(3))) float lds_float_t;

#define FEATS     64
#define ROWB      288                    // padded LDS row stride (72 dwords) ->
#define ROWF      72                     //   conflict-free strided B reads
#define BATCH_OFF (4 * ROWB)             // 16B of batch ids per slot
#define SLOTB     (4 * ROWB + 16)        // 4-node tile + batch ids = 1168 B
#define DEPTH     6                      // async pipeline depth (tiles)
#define WPB       8                      // waves per block
#define BLOCK     (WPB * 32)
#define WAVE_LDS_B (DEPTH * SLOTB)       // 7008 B per wave (56,064 B / block)

__device__ __forceinline__ v8f vzero8() {
    v8f v = {0.f, 0.f, 0.f, 0.f, 0.f, 0.f, 0.f, 0.f};
    return v;
}

__global__ __launch_bounds__(BLOCK)
void pool_kernel(const float* __restrict__ x, const int* __restrict__ batch,
                 float* __restrict__ out, float* __restrict__ counts,
                 long total_steps, int num_waves, long n_nodes)
{
    __shared__ __align__(16) unsigned char smem[WPB * WAVE_LDS_B];

    const int lane  = threadIdx.x & 31;
    const int wslot = threadIdx.x >> 5;
    const int wid   = blockIdx.x * WPB + wslot;
    const int row   = lane & 15;     // A: M index / B,C,D: N index
    const int hi    = lane >> 4;     // K half-select (K = 2*hi, 2*hi+1)

    const long s0 = (total_steps * (long)wid) / num_waves;
    const long s1 = (total_steps * (long)(wid + 1)) / num_waves;

    unsigned char* wbase = smem + wslot * WAVE_LDS_B;
    // Raw LDS byte address of this wave's staging region (HW adds LDS_BASE).
    const uint32_t lds0 = (uint32_t)(uintptr_t)(lds_float_t*)wbase;

    v8f acc0 = vzero8(), acc1 = vzero8(), acc2 = vzero8(), acc3 = vzero8();
    float cnt = 0.f;
    int g_base = 0;                  // scalar (readfirstlane'd) window base

    // Stage one 4-node tile (1 KB features + 16 B batch ids) asynchronously.
    auto issue = [&](long s, int slot) {
        uint64_t g = (uint64_t)(uintptr_t)x
                   + ((uint64_t)s * 4u + (uint64_t)hi) * (FEATS * 4)
                   + (uint64_t)row * 16;
        uint32_t l = lds0 + (uint32_t)slot * SLOTB + (uint32_t)hi * ROWB
                   + (uint32_t)row * 16;
        asm volatile("global_load_async_to_lds_b128 %0, %1, off"
                     :: "v"(l), "v"(g) : "memory");
        g += 2u * FEATS * 4;         // rows n+2, n+3
        l += 2u * ROWB;
        asm volatile("global_load_async_to_lds_b128 %0, %1, off"
                     :: "v"(l), "v"(g) : "memory");
        // batch[s*4 .. s*4+3]: lanes 0..3 carry one dword each
        uint64_t gb = (uint64_t)(uintptr_t)batch + (uint64_t)s * 16
                    + (uint32_t)lane * 4;
        uint32_t lb = lds0 + (uint32_t)slot * SLOTB + BATCH_OFF
                    + (uint32_t)lane * 4;
        if (lane < 4) {
            asm volatile("global_load_async_to_lds_b32 %0, %1, off"
                         :: "v"(lb), "v"(gb) : "memory");
        }
    };

    // Flush 16-graph x 64-feature accumulator window (skip exact zeros).
    auto flush = [&]() {
        #pragma unroll
        for (int j = 0; j < 8; ++j) {
            const int g = g_base + j + 8 * hi;       // C/D: M = j (+8 hi half)
            float* op = out + g * FEATS + row;
            if (acc0[j] != 0.f) atomicAdd(op +  0, acc0[j]);
            if (acc1[j] != 0.f) atomicAdd(op + 16, acc1[j]);
            if (acc2[j] != 0.f) atomicAdd(op + 32, acc2[j]);
            if (acc3[j] != 0.f) atomicAdd(op + 48, acc3[j]);
        }
        if (cnt != 0.f) atomicAdd(counts + g_base + row, cnt);
        acc0 = vzero8(); acc1 = vzero8(); acc2 = vzero8(); acc3 = vzero8();
        cnt = 0.f;
    };

    // Consume one staged tile: one-hot A vs 16-graph window, 4 WMMAs.
    auto consume = [&](int slot) {
        const int4 b4 = *(const int4*)(wbase + slot * SLOTB + BATCH_OFF);
        const int bw = __builtin_amdgcn_readfirstlane(b4.w);
        const int bx = __builtin_amdgcn_readfirstlane(b4.x);
        if (bw >= g_base + 16) {     // scalar branch: EXEC stays all-1s
            flush();
            g_base = bx;
        }
        const int gA = hi ? b4.z : b4.x;   // graph of node K=2*hi
        const int gB = hi ? b4.w : b4.y;   // graph of node K=2*hi+1
        v2f a;
        a.x = (gA == g_base + row) ? 1.f : 0.f;
        a.y = (gB == g_base + row) ? 1.f : 0.f;
        cnt += a.x + a.y;

        const float* sp = (const float*)(wbase + slot * SLOTB)
                        + (2 * hi) * ROWF + row;
        v2f b0 = { sp[ 0], sp[ROWF +  0] };
        v2f b1 = { sp[16], sp[ROWF + 16] };
        v2f b2 = { sp[32], sp[ROWF + 32] };
        v2f b3 = { sp[48], sp[ROWF + 48] };

        acc0 = __builtin_amdgcn_wmma_f32_16x16x4_f32(false, a, false, b0, (short)0, acc0, false, false);
        acc1 = __builtin_amdgcn_wmma_f32_16x16x4_f32(false, a, false, b1, (short)0, acc1, false, false);
        acc2 = __builtin_amdgcn_wmma_f32_16x16x4_f32(false, a, false, b2, (short)0, acc2, false, false);
        acc3 = __builtin_amdgcn_wmma_f32_16x16x4_f32(false, a, false, b3, (short)0, acc3, false, false);
    };

    if (s0 < s1) {
        g_base = __builtin_amdgcn_readfirstlane(batch[s0 * 4]);

        // Prologue: fill the async pipeline.
        int nIss = 0;
        for (; nIss < DEPTH && (s0 + nIss) < s1; ++nIss) issue(s0 + nIss, nIss);

        int slot = 0;
        const long sSteady = (s1 - s0 > DEPTH) ? (s1 - DEPTH) : s0;

        // Steady state: 18 async transfers outstanding per wave (6 KB).
        for (long s = s0; s < sSteady; ++s) {
            asm volatile("s_wait_asynccnt 0xf" ::: "memory"); // oldest tile done
            consume(slot);
            issue(s + DEPTH, slot);
            slot = (slot + 1 == DEPTH) ? 0 : slot + 1;
        }
        // Drain.
        asm volatile("s_wait_asynccnt 0x0" ::: "memory");
        for (long s = sSteady; s < s1; ++s) {
            consume(slot);
            slot = (slot + 1 == DEPTH) ? 0 : slot + 1;
        }
        flush();
    }

    // Tail nodes (n_nodes % 4 != 0): scalar-atomic path, last wave only.
    if (wid == num_waves - 1) {
        for (long n = total_steps * 4; n < n_nodes; ++n) {
            const int g = batch[n];
            const int f = lane * 2;
            atomicAdd(out + g * FEATS + f,     x[n * FEATS + f]);
            atomicAdd(out + g * FEATS + f + 1, x[n * FEATS + f + 1]);
            if (lane == 0) atomicAdd(counts + g, 1.f);
        }
    }
}

__global__ void zero_kernel(float* __restrict__ out, float* __restrict__ counts,
                            int on, int gn)
{
    const int i = blockIdx.x * blockDim.x + threadIdx.x;
    if (i < on) out[i] = 0.f;
    if (i < gn) counts[i] = 0.f;
}

__global__ void finalize_kernel(float* __restrict__ out,
                                const float* __restrict__ counts, int on)
{
    const int i = blockIdx.x * blockDim.x + threadIdx.x;
    if (i < on) {
        const float c = counts[i >> 6];   // FEATS == 64
        const float s = out[i];
        out[i] = (c > 0.f) ? (s / c) : 0.f;
    }
}

extern "C" void kernel_launch(void* const* d_in, const int* in_sizes, int n_in,
                              void* d_out, int out_size, void* d_ws, size_t ws_size,
                              hipStream_t stream)
{
    const float* x      = (const float*)d_in[0];
    const int*   batch  = (const int*)d_in[1];
    float*       out    = (float*)d_out;
    float*       counts = (float*)d_ws;          // G floats of scratch

    const long n_nodes = (long)in_sizes[1];
    const int  G       = out_size / FEATS;

    const int zn = (out_size > G) ? out_size : G;
    zero_kernel<<<(zn + 255) / 256, 256, 0, stream>>>(out, counts, out_size, G);

    const long steps    = n_nodes / 4;
    const int  blocks   = 488;                   // 3904 waves: ~23MB in flight
    const int  numWaves = blocks * WPB;
    pool_kernel<<<blocks, BLOCK, 0, stream>>>(x, batch, out, counts,
                                              steps, numWaves, n_nodes);

    finalize_kernel<<<(out_size + 255) / 256, 256, 0, stream>>>(out, counts, out_size);
}